// Kmeans_64450279244052
// MI455X (gfx1250) — compile-verified
//
#include <hip/hip_runtime.h>
#include <math.h>

typedef __attribute__((ext_vector_type(2))) float    v2f;
typedef __attribute__((ext_vector_type(8))) float    v8f;
typedef __attribute__((ext_vector_type(4))) unsigned ui4;
typedef __attribute__((ext_vector_type(8))) unsigned ui8;

#define B_ 4
#define H_ 8
#define L_ 8192
#define D_ 64
#define C_ 256
#define W_ 32
#define COMMIT 1e-4f

// workspace layout (in 4-byte words)
#define DT_ELEMS  ((size_t)B_ * H_ * C_ * L_)            // transposed dists [bh][c][l]
#define SUMS_OFF  (DT_ELEMS)                              // H_*C_*D_ floats
#define BINS_OFF  (SUMS_OFF + (size_t)H_ * C_ * D_)       // H_*C_ uints
#define LOSS_OFF  (BINS_OFF + (size_t)H_ * C_)            // 1 float
#define WS_WORDS  (LOSS_OFF + 1)

// output layout: indices [B_*H_*C_*W_] as float, then loss [1], then new_means [H_*C_*D_]
#define OUT_LOSS_OFF  ((size_t)B_ * H_ * C_ * W_)
#define OUT_MEANS_OFF (OUT_LOSS_OFF + 1)

__global__ __launch_bounds__(256) void init_ws_kernel(unsigned* __restrict__ w, unsigned nwords) {
    unsigned i = blockIdx.x * 256u + threadIdx.x;
    if (i < nwords) w[i] = 0u;
}

// Fused: TDM means->LDS + dists GEMM (f32 WMMA) + LDS-transposed coalesced dist
// store + argmax buckets + cluster scatter-add + commitment-loss accumulation.
// grid = (L_/128, B_*H_), block = 256 (8 waves; each wave owns 16 rows of l).
__global__ __launch_bounds__(256) void kmeans_main_kernel(
    const float* __restrict__ x, const float* __restrict__ means,
    float* __restrict__ dt, float* __restrict__ sums,
    unsigned* __restrict__ bins, float* __restrict__ lossAcc)
{
    __shared__ float sm[C_ * D_];           // 64 KB: means slice for this h
    __shared__ float stage[8][16 * 17];     // per-wave 16x16 transpose buffer, stride 17

    const int tid  = threadIdx.x;
    const int bh   = blockIdx.y;
    const int h    = bh & (H_ - 1);
    const int wave = tid >> 5;
    const int lane = tid & 31;

    // ---- Tensor Data Mover: DMA means[h] (256 x 64 f32, contiguous) into LDS ----
    if (wave == 0) {
        const unsigned lds_off = (unsigned)(size_t)(&sm[0]);      // LDS aperture: low 32 bits = byte offset
        const unsigned long long ga = (unsigned long long)(means + (size_t)h * C_ * D_);
        // D# group 0: count=1 | lds_addr | global_addr[56:0] | type=2
        ui4 g0;
        g0.x = 1u;
        g0.y = lds_off;
        g0.z = (unsigned)ga;
        g0.w = (unsigned)((ga >> 32) & 0x1FFFFFFull) | (2u << 30);
        // D# group 1: data_size=4B; tensor_dim0=64, tensor_dim1=256;
        //             tile_dim0=64, tile_dim1=256; tensor_dim0_stride=64
        ui8 g1;
        g1.s0 = (2u << 16);                               // wg_mask=0, data_size=2 (4B)
        g1.s1 = (unsigned)(D_ & 0xFFFF) << 16;            // tensor_dim0[15:0] @ bit48
        g1.s2 = ((unsigned)D_ >> 16) | ((unsigned)(C_ & 0xFFFF) << 16); // dim0 hi | tensor_dim1 lo
        g1.s3 = ((unsigned)C_ >> 16) | ((unsigned)D_ << 16);            // dim1 hi | tile_dim0
        g1.s4 = (unsigned)C_;                             // tile_dim1=256, tile_dim2=0
        g1.s5 = (unsigned)D_;                             // tensor_dim0_stride = 64 (low 32)
        g1.s6 = 0u;
        g1.s7 = 0u;
        asm volatile("tensor_load_to_lds %0, %1" : : "s"(g0), "s"(g1) : "memory");
        __builtin_amdgcn_s_wait_tensorcnt(0);
    }
    __syncthreads();

    const int half = lane >> 4;     // which K-pair half of the wave
    const int lm   = lane & 15;
    const int rowbase = blockIdx.x * 128 + wave * 16;
    const float* xb = x + (size_t)bh * L_ * D_;

    // A fragments for all 16 K-steps (16x4 f32 layout: M = lane&15, K pair by half)
    v2f afr[16];
    {
        const float* xr = xb + (size_t)(rowbase + lm) * D_ + 2 * half;
        #pragma unroll
        for (int i = 0; i < 16; ++i) {
            const float* p = xr + 4 * i;
            v2f a; a[0] = p[0]; a[1] = p[1];
            afr[i] = a;
        }
    }

    float bestv[8];
    int   besti[8];
    #pragma unroll
    for (int r = 0; r < 8; ++r) { bestv[r] = -INFINITY; besti[r] = 0; }

    float* st = &stage[wave][0];

    // 16 N-tiles of 16 columns cover c=256
    for (int nt = 0; nt < 16; ++nt) {
        const int n = nt * 16 + lm;                 // this lane's column
        const float* mrow = sm + n * D_ + 2 * half; // B[k][n] = means[n][k]
        v8f acc = {};
        #pragma unroll
        for (int i = 0; i < 16; ++i) {
            const float* p = mrow + 4 * i;
            v2f bfr; bfr[0] = p[0]; bfr[1] = p[1];
            acc = __builtin_amdgcn_wmma_f32_16x16x4_f32(
                false, afr[i], false, bfr, (short)0, acc, false, false);
        }
        // running per-row argmax (C/D layout: VGPR r -> row r+8*half, col lane&15)
        #pragma unroll
        for (int r = 0; r < 8; ++r) {
            const float v = acc[r];
            if (v > bestv[r]) { bestv[r] = v; besti[r] = n; }  // strict > keeps lowest n
        }
        // transpose tile through wave-private LDS (stride 17 = conflict-free),
        // then coalesced b128 stores: 4 lanes cover a 64B run per cluster row
        #pragma unroll
        for (int r = 0; r < 8; ++r)
            st[lm * 17 + 8 * half + r] = acc[r];
        #pragma unroll
        for (int p = 0; p < 2; ++p) {
            const int n_loc = p * 8 + (lane >> 2);
            const int l4    = (lane & 3) * 4;
            const float* sp = st + n_loc * 17 + l4;
            float4 q; q.x = sp[0]; q.y = sp[1]; q.z = sp[2]; q.w = sp[3];
            *(float4*)(dt + ((size_t)bh * C_ + nt * 16 + n_loc) * L_ + rowbase + l4) = q;
        }
    }

    // per-row argmax: butterfly reduce across the 16 lanes of each half-group
    #pragma unroll
    for (int r = 0; r < 8; ++r) {
        float v = bestv[r]; int idx = besti[r];
        #pragma unroll
        for (int m = 8; m >= 1; m >>= 1) {
            float ov = __shfl_xor(v, m, 32);
            int   oi = __shfl_xor(idx, m, 32);
            if (ov > v || (ov == v && oi < idx)) { v = ov; idx = oi; }
        }
        bestv[r] = v; besti[r] = idx;
    }

    // broadcast bucket of each of the wave's 16 rows to all lanes
    int bk[16];
    #pragma unroll
    for (int r = 0; r < 8; ++r) {
        bk[r]     = __shfl(besti[r], 0, 32);   // rows 0..7  live in lanes 0-15
        bk[r + 8] = __shfl(besti[r], 16, 32);  // rows 8..15 live in lanes 16-31
    }

    // scatter-add into cluster sums/bins + commitment loss (routed to ORIGINAL means)
    float lossLocal = 0.f;
    #pragma unroll
    for (int rr = 0; rr < 16; ++rr) {
        const int rglob = rowbase + rr;
        const int bkt   = bk[rr];
        const float* xr = xb + (size_t)rglob * D_;
        const float x0 = xr[lane], x1 = xr[lane + 32];
        const float m0 = sm[bkt * D_ + lane], m1 = sm[bkt * D_ + lane + 32];
        const float d0 = x0 - m0, d1 = x1 - m1;
        lossLocal += d0 * d0 + d1 * d1;
        float* srow = sums + ((size_t)h * C_ + bkt) * D_;
        atomicAdd(&srow[lane], x0);
        atomicAdd(&srow[lane + 32], x1);
        if (lane == 0) atomicAdd(&bins[h * C_ + bkt], 1u);
    }
    #pragma unroll
    for (int m = 16; m >= 1; m >>= 1) lossLocal += __shfl_xor(lossLocal, m, 32);
    if (lane == 0) atomicAdd(lossAcc, lossLocal);
}

// grid = H_*C_ blocks of 64 threads: L2-normalize sums, keep old mean if bin empty.
__global__ __launch_bounds__(64) void finalize_means_kernel(
    const float* __restrict__ means, const float* __restrict__ sums,
    const unsigned* __restrict__ bins, const float* __restrict__ lossAcc,
    float* __restrict__ out)
{
    __shared__ float red[64];
    const int hc = blockIdx.x;
    const int t  = threadIdx.x;
    const float s = sums[(size_t)hc * D_ + t];
    red[t] = s * s;
    __syncthreads();
    for (int st = 32; st >= 1; st >>= 1) {
        if (t < st) red[t] += red[t + st];
        __syncthreads();
    }
    const float denom = fmaxf(sqrtf(red[0]), 1e-12f);
    const float outv = (bins[hc] == 0u) ? means[(size_t)hc * D_ + t] : (s / denom);
    out[OUT_MEANS_OFF + (size_t)hc * D_ + t] = outv;
    if (hc == 0 && t == 0)
        out[OUT_LOSS_OFF] = lossAcc[0] * (1.0f / ((float)B_ * H_ * L_ * D_)) * COMMIT;
}

// grid = B_*H_*C_ blocks of 256 threads: top-32 of 8192, indices sorted ascending.
__global__ __launch_bounds__(256) void topk_kernel(
    const float* __restrict__ dt, float* __restrict__ out)
{
    __shared__ float vals[L_];     // 32 KB
    __shared__ float rv[256];
    __shared__ int   ri[256];
    __shared__ int   topi[W_];
    const int t = threadIdx.x;
    const size_t row = blockIdx.x;
    const float* src = dt + row * L_;
    for (int i = t; i < L_; i += 256) vals[i] = src[i];
    __syncthreads();

    for (int sel = 0; sel < W_; ++sel) {
        float bv = -INFINITY; int bi = L_;
        for (int i = t; i < L_; i += 256) {
            const float v = vals[i];
            if (v > bv) { bv = v; bi = i; }     // strict > keeps lowest index
        }
        rv[t] = bv; ri[t] = bi;
        __syncthreads();
        for (int st = 128; st >= 1; st >>= 1) {
            if (t < st) {
                if (rv[t + st] > rv[t] || (rv[t + st] == rv[t] && ri[t + st] < ri[t])) {
                    rv[t] = rv[t + st]; ri[t] = ri[t + st];
                }
            }
            __syncthreads();
        }
        if (t == 0) { topi[sel] = ri[0]; vals[ri[0]] = -INFINITY; }
        __syncthreads();
    }

    if (t == 0) {   // ascending insertion sort of 32 indices
        for (int i = 1; i < W_; ++i) {
            int key = topi[i]; int j = i - 1;
            while (j >= 0 && topi[j] > key) { topi[j + 1] = topi[j]; --j; }
            topi[j + 1] = key;
        }
    }
    __syncthreads();
    if (t < W_) out[row * W_ + t] = (float)topi[t];
}

extern "C" void kernel_launch(void* const* d_in, const int* in_sizes, int n_in,
                              void* d_out, int out_size, void* d_ws, size_t ws_size,
                              hipStream_t stream) {
    const float* x     = (const float*)d_in[0];
    const float* means = (const float*)d_in[1];
    // d_in[2] = window_size scalar (compile-time W_=32 here)
    float* out = (float*)d_out;

    float*    ws      = (float*)d_ws;
    float*    dt      = ws;
    float*    sums    = ws + SUMS_OFF;
    unsigned* bins    = (unsigned*)(ws + BINS_OFF);
    float*    lossAcc = ws + LOSS_OFF;

    // zero accumulators (sums + bins + loss) every call — graph-replay safe
    const unsigned nz = (unsigned)(WS_WORDS - SUMS_OFF);
    init_ws_kernel<<<(nz + 255) / 256, 256, 0, stream>>>((unsigned*)sums, nz);

    dim3 grid(L_ / 128, B_ * H_);
    kmeans_main_kernel<<<grid, 256, 0, stream>>>(x, means, dt, sums, bins, lossAcc);

    finalize_means_kernel<<<H_ * C_, 64, 0, stream>>>(means, sums, bins, lossAcc, out);

    topk_kernel<<<B_ * H_ * C_, 256, 0, stream>>>(dt, out);
}